// GCNII_17626545783193
// MI455X (gfx1250) — compile-verified
//
#include <hip/hip_runtime.h>
#include <math.h>

// GCNII forward for MI455X (gfx1250, wave32).
// Dense GEMMs use V_WMMA_F32_16X16X4_F32 (fp32-exact, matches reference math).
// SpMM uses L2-resident float atomics: all state (~78MB) fits in the 192MB L2.

typedef __attribute__((ext_vector_type(2))) float v2f;
typedef __attribute__((ext_vector_type(8))) float v8f;

#define NFEATS   128
#define HIDDEN   64
#define NCLASSES 40
#define NLAYERS  8

// ---------------- degree / symmetric normalization ----------------

__global__ void k_deg_init(float* __restrict__ deg, int n) {
    int i = blockIdx.x * blockDim.x + threadIdx.x;
    if (i < n) deg[i] = 1.0f;                 // self-loop weight
}

__global__ void k_deg_edge(const int* __restrict__ col, float* __restrict__ deg, int nE) {
    int e = blockIdx.x * blockDim.x + threadIdx.x;
    if (e < nE) atomicAdd(&deg[col[e]], 1.0f);
}

__global__ void k_rsqrt_inplace(float* __restrict__ deg, int n) {
    int i = blockIdx.x * blockDim.x + threadIdx.x;
    if (i < n) { float d = deg[i]; deg[i] = d > 0.0f ? rsqrtf(d) : 0.0f; }
}

// ---------------- SpMM: agg[t] += dinv[s]*dinv[t] * h[s] ----------------

// agg[i][:] = dinv[i]^2 * h[i][:]   (self-loop term; also zero-initializes agg)
__global__ void k_agg_init(const float* __restrict__ dinv, const float* __restrict__ h,
                           float* __restrict__ agg, int total) {
    int i = blockIdx.x * blockDim.x + threadIdx.x;
    if (i < total) {
        float d = dinv[i >> 6];               // node = i / HIDDEN
        agg[i] = d * d * h[i];
    }
}

// 16 threads per edge, 4 consecutive features each (float4 gather, 4 f32 atomics).
__global__ void k_spmm(const int* __restrict__ row, const int* __restrict__ col,
                       const float* __restrict__ dinv, const float* __restrict__ h,
                       float* __restrict__ agg, int nE) {
    unsigned tid = blockIdx.x * blockDim.x + threadIdx.x;
    int e = tid >> 4;
    if (e >= nE) return;
    int f = (tid & 15) * 4;
    int s = row[e], t = col[e];
    float w = dinv[s] * dinv[t];
    const float4 hv = *(const float4*)(h + (size_t)s * HIDDEN + f);
    float* dst = agg + (size_t)t * HIDDEN + f;
    atomicAdd(dst + 0, w * hv.x);
    atomicAdd(dst + 1, w * hv.y);
    atomicAdd(dst + 2, w * hv.z);
    atomicAdd(dst + 3, w * hv.w);
}

// ---------------- WMMA fragment helpers (V_WMMA_F32_16X16X4_F32) ----------------
// A 16x4 (MxK): lanes 0-15 M=0..15 (K=kk,kk+1), lanes 16-31 M=0..15 (K=kk+2,kk+3).
// B 4x16 (KxN): lane%16 = N; VGPR0 = rows {kk, kk+2}, VGPR1 = rows {kk+1, kk+3}
//               (lane-half selects which of the pair).
// C/D 16x16:    VGPR v: lanes 0-15 -> M=v, lanes 16-31 -> M=v+8; N = lane%16.

__device__ __forceinline__ v2f bfrag_rowmajor(const float* __restrict__ B, int ldb,
                                              int kb /* kk + lanehalf*2 */, int ncol) {
    v2f b;
    b.x = B[(size_t)kb * ldb + ncol];
    b.y = B[(size_t)(kb + 1) * ldb + ncol];
    return b;
}

#define WMMA_F32X4(acc, a, b) \
    (acc) = __builtin_amdgcn_wmma_f32_16x16x4_f32(false, (a), false, (b), (short)0, (acc), false, false)

// ---------------- h = relu(x @ W0 + b0);  h0 = h ----------------

__global__ __launch_bounds__(256) void k_gemm0(const float* __restrict__ x,
                                               const float* __restrict__ W0,
                                               const float* __restrict__ b0,
                                               float* __restrict__ h,
                                               float* __restrict__ h0, int n) {
    const int lane = threadIdx.x & 31;
    const int wave = threadIdx.x >> 5;
    const int m0   = (blockIdx.x * 8 + wave) * 16;
    const int ml   = lane & 15;
    const int kh   = lane >> 4;                 // lane half: 0 or 1
    int m = m0 + ml; if (m >= n) m = n - 1;     // clamp loads, mask stores

    v8f acc[4] = {};
    for (int kk = 0; kk < NFEATS; kk += 4) {
        const int ka = kk + kh * 2;
        const float2 av = *(const float2*)(x + (size_t)m * NFEATS + ka);
        v2f a; a.x = av.x; a.y = av.y;
        #pragma unroll
        for (int nt = 0; nt < 4; ++nt) {
            v2f b = bfrag_rowmajor(W0, HIDDEN, ka, nt * 16 + ml);
            WMMA_F32X4(acc[nt], a, b);
        }
    }
    const int rbase = m0 + kh * 8;
    const bool full = (m0 + 16 <= n);           // wave-uniform fast path
    #pragma unroll
    for (int nt = 0; nt < 4; ++nt) {
        int c = nt * 16 + ml;
        float bias = b0[c];
        if (full) {
            #pragma unroll
            for (int v = 0; v < 8; ++v) {
                float val = fmaxf(acc[nt][v] + bias, 0.0f);
                size_t o = (size_t)(rbase + v) * HIDDEN + c;
                h[o] = val; h0[o] = val;
            }
        } else {
            #pragma unroll
            for (int v = 0; v < 8; ++v) {
                int r = rbase + v;
                if (r < n) {
                    float val = fmaxf(acc[nt][v] + bias, 0.0f);
                    size_t o = (size_t)r * HIDDEN + c;
                    h[o] = val; h0[o] = val;
                }
            }
        }
    }
}

// -------- h = relu(beta*(support @ W) + (1-beta)*support), support = 0.9*agg + 0.1*h0 --------

__global__ __launch_bounds__(256) void k_layer(const float* __restrict__ agg,
                                               const float* __restrict__ h0,
                                               const float* __restrict__ W,
                                               float* __restrict__ h,
                                               float beta, int n) {
    const int lane = threadIdx.x & 31;
    const int wave = threadIdx.x >> 5;
    const int m0   = (blockIdx.x * 8 + wave) * 16;
    const int ml   = lane & 15;
    const int kh   = lane >> 4;
    int m = m0 + ml; if (m >= n) m = n - 1;

    v8f acc[4] = {};
    for (int kk = 0; kk < HIDDEN; kk += 4) {
        const int ka = kk + kh * 2;
        size_t idx = (size_t)m * HIDDEN + ka;
        const float2 av = *(const float2*)(agg + idx);
        const float2 hv = *(const float2*)(h0 + idx);
        v2f a;
        a.x = 0.9f * av.x + 0.1f * hv.x;
        a.y = 0.9f * av.y + 0.1f * hv.y;
        #pragma unroll
        for (int nt = 0; nt < 4; ++nt) {
            v2f b = bfrag_rowmajor(W, HIDDEN, ka, nt * 16 + ml);
            WMMA_F32X4(acc[nt], a, b);
        }
    }
    const float omb = 1.0f - beta;
    const int rbase = m0 + kh * 8;
    const bool full = (m0 + 16 <= n);
    #pragma unroll
    for (int nt = 0; nt < 4; ++nt) {
        int c = nt * 16 + ml;
        if (full) {
            #pragma unroll
            for (int v = 0; v < 8; ++v) {
                size_t o = (size_t)(rbase + v) * HIDDEN + c;
                float sup = 0.9f * agg[o] + 0.1f * h0[o];
                h[o] = fmaxf(beta * acc[nt][v] + omb * sup, 0.0f);
            }
        } else {
            #pragma unroll
            for (int v = 0; v < 8; ++v) {
                int r = rbase + v;
                if (r < n) {
                    size_t o = (size_t)r * HIDDEN + c;
                    float sup = 0.9f * agg[o] + 0.1f * h0[o];
                    h[o] = fmaxf(beta * acc[nt][v] + omb * sup, 0.0f);
                }
            }
        }
    }
}

// ---------------- out = h @ W_out + b_out  (N=40, 3 col-tiles, masked) ----------------

__global__ __launch_bounds__(256) void k_out(const float* __restrict__ h,
                                             const float* __restrict__ Wout,
                                             const float* __restrict__ bout,
                                             float* __restrict__ out, int n) {
    const int lane = threadIdx.x & 31;
    const int wave = threadIdx.x >> 5;
    const int m0   = (blockIdx.x * 8 + wave) * 16;
    const int ml   = lane & 15;
    const int kh   = lane >> 4;
    int m = m0 + ml; if (m >= n) m = n - 1;

    v8f acc[3] = {};
    for (int kk = 0; kk < HIDDEN; kk += 4) {
        const int ka = kk + kh * 2;
        const float2 av = *(const float2*)(h + (size_t)m * HIDDEN + ka);
        v2f a; a.x = av.x; a.y = av.y;
        #pragma unroll
        for (int nt = 0; nt < 3; ++nt) {
            int c = nt * 16 + ml;
            v2f b;
            b.x = (c < NCLASSES) ? Wout[(size_t)ka * NCLASSES + c] : 0.0f;
            b.y = (c < NCLASSES) ? Wout[(size_t)(ka + 1) * NCLASSES + c] : 0.0f;
            WMMA_F32X4(acc[nt], a, b);
        }
    }
    const int rbase = m0 + kh * 8;
    const bool full = (m0 + 16 <= n);
    #pragma unroll
    for (int nt = 0; nt < 3; ++nt) {
        int c = nt * 16 + ml;
        if (c < NCLASSES) {
            float bias = bout[c];
            if (full) {
                #pragma unroll
                for (int v = 0; v < 8; ++v)
                    out[(size_t)(rbase + v) * NCLASSES + c] = acc[nt][v] + bias;
            } else {
                #pragma unroll
                for (int v = 0; v < 8; ++v) {
                    int r = rbase + v;
                    if (r < n) out[(size_t)r * NCLASSES + c] = acc[nt][v] + bias;
                }
            }
        }
    }
}

// ---------------- host-side orchestration ----------------

extern "C" void kernel_launch(void* const* d_in, const int* in_sizes, int n_in,
                              void* d_out, int out_size, void* d_ws, size_t ws_size,
                              hipStream_t stream) {
    const float* x    = (const float*)d_in[0];
    const int*   ei   = (const int*)  d_in[1];   // [2, E]
    const float* W0   = (const float*)d_in[2];
    const float* b0   = (const float*)d_in[3];
    const float* Wl   = (const float*)d_in[4];   // [NLAYERS, 64, 64]
    const float* Wout = (const float*)d_in[5];
    const float* bout = (const float*)d_in[6];

    const int N = in_sizes[0] / NFEATS;
    const int E = in_sizes[1] / 2;
    const int* erow = ei;          // edge_index[0] = source
    const int* ecol = ei + E;      // edge_index[1] = target

    // workspace: deg/dinv (N, rounded), h, h0, agg (N*64 each) ~= 78 MB
    float* deg = (float*)d_ws;
    size_t nAl = ((size_t)N + 63) & ~(size_t)63;
    float* h   = deg + nAl;
    float* h0  = h  + (size_t)N * HIDDEN;
    float* agg = h0 + (size_t)N * HIDDEN;
    (void)ws_size; (void)n_in; (void)out_size;

    const int nodeBlocks = (N + 255) / 256;
    const int edgeBlocks = (E + 255) / 256;
    const int gemmBlocks = (N + 127) / 128;               // 8 waves x 16 rows per block
    const int total      = N * HIDDEN;
    const int aggBlocks  = (total + 255) / 256;
    const long spmmThr   = (long)E * 16;
    const int spmmBlocks = (int)((spmmThr + 255) / 256);

    k_deg_init     <<<nodeBlocks, 256, 0, stream>>>(deg, N);
    k_deg_edge     <<<edgeBlocks, 256, 0, stream>>>(ecol, deg, E);
    k_rsqrt_inplace<<<nodeBlocks, 256, 0, stream>>>(deg, N);   // deg now holds dinv

    k_gemm0<<<gemmBlocks, 256, 0, stream>>>(x, W0, b0, h, h0, N);

    for (int i = 0; i < NLAYERS; ++i) {
        k_agg_init<<<aggBlocks,  256, 0, stream>>>(deg, h, agg, total);
        k_spmm    <<<spmmBlocks, 256, 0, stream>>>(erow, ecol, deg, h, agg, E);
        float beta = logf(0.5f / (float)(i + 1) + 1.0f);
        k_layer<<<gemmBlocks, 256, 0, stream>>>(agg, h0, Wl + (size_t)i * HIDDEN * HIDDEN,
                                                h, beta, N);
    }

    k_out<<<gemmBlocks, 256, 0, stream>>>(h, Wout, bout, (float*)d_out, N);
}